// CoSenCrossEntropyLoss_22290880266922
// MI455X (gfx1250) — compile-verified
//
#include <hip/hip_runtime.h>
#include <hip/hip_bf16.h>
#include <cstdint>
#include <cstddef>

// CoSen cross-entropy loss, MI455X (gfx1250, wave32).
// Memory-bound reduction: 131 MB of cls_score -> ~5.6us HBM floor @ 23.3 TB/s.
// CDNA5 paths used: global_load_async_to_lds_b128 (ASYNCcnt DMA staging),
// s_wait_asynccnt, wave32 shuffle butterflies. WMMA is intentionally NOT used:
// there is no contraction against a dense operand (row-reductions on a ones
// vector would waste 15/16 of every 16x16 WMMA tile).

#define B_DIM 32768
#define C_DIM 1000
#define ROWS_PER_BLOCK 8
#define THREADS (ROWS_PER_BLOCK * 32)   // 8 waves of 32
#define NBLOCKS (B_DIM / ROWS_PER_BLOCK)

// Builtin signature (revealed by clang diagnostic): pointee type is a 16-byte
// int vector, global side in addrspace(1), LDS side in addrspace(3).
typedef int v4i __attribute__((ext_vector_type(4)));
typedef __attribute__((address_space(1))) v4i* g_v4i_ptr;
typedef __attribute__((address_space(3))) v4i* l_v4i_ptr;

#if __has_builtin(__builtin_amdgcn_global_load_async_to_lds_b128)
#define HAVE_ASYNC_LDS 1
#else
#define HAVE_ASYNC_LDS 0
#endif

static __device__ __forceinline__ void wait_async_zero() {
#if __has_builtin(__builtin_amdgcn_s_wait_asynccnt)
    __builtin_amdgcn_s_wait_asynccnt(0);
#else
    asm volatile("s_wait_asynccnt 0" ::: "memory");
#endif
}

__global__ __launch_bounds__(THREADS) void cosen_row_kernel(
    const float* __restrict__ cls,        // [B, C]
    const int*   __restrict__ label,      // [B]
    const float* __restrict__ xi,         // [C, C]
    float*       __restrict__ block_part) // [NBLOCKS]
{
    __shared__ float srows[ROWS_PER_BLOCK][C_DIM];   // 32000 B
    __shared__ float vrow[ROWS_PER_BLOCK];

    const int lane = threadIdx.x & 31;
    const int wid  = threadIdx.x >> 5;
    const int row  = blockIdx.x * ROWS_PER_BLOCK + wid;

    const float* grow = cls + (size_t)row * C_DIM;
    float* lrow = srows[wid];

    // ---- Stage this wave's row into LDS (async DMA, ASYNCcnt) ---------
#if HAVE_ASYNC_LDS
    constexpr int NXFER = (C_DIM * 4) / 16;          // 250 x B128 transfers
    #pragma unroll
    for (int base = 0; base < NXFER; base += 32) {
        const int idx = base + lane;
        if (idx < NXFER) {
            __builtin_amdgcn_global_load_async_to_lds_b128(
                (g_v4i_ptr)(grow + idx * 4),
                (l_v4i_ptr)(lrow + idx * 4),
                /*offset=*/0, /*cpol=*/0);
        }
    }
    wait_async_zero();   // wave reads only the LDS it filled itself
#else
    {
        const float4* g4 = (const float4*)grow;
        float4* l4w = (float4*)lrow;
        for (int i = lane; i < C_DIM / 4; i += 32) l4w[i] = g4[i];
    }
#endif

    // ---- Pass 1: max + argmax (first-max tie-break, like jnp.argmax) ---
    float vmax = -3.402823466e38f;
    int   imax = 0;
    const float4* l4 = (const float4*)lrow;
    for (int i = lane; i < C_DIM / 4; i += 32) {
        const float4 v = l4[i];
        const int j = i * 4;
        if (v.x > vmax) { vmax = v.x; imax = j;     }
        if (v.y > vmax) { vmax = v.y; imax = j + 1; }
        if (v.z > vmax) { vmax = v.z; imax = j + 2; }
        if (v.w > vmax) { vmax = v.w; imax = j + 3; }
    }
    #pragma unroll
    for (int off = 16; off > 0; off >>= 1) {
        const float ov = __shfl_xor(vmax, off, 32);
        const int   oi = __shfl_xor(imax, off, 32);
        if (ov > vmax || (ov == vmax && oi < imax)) { vmax = ov; imax = oi; }
    }

    // ---- Pass 2: sum_j exp(s_ij - vmax) (from LDS) ---------------------
    float acc = 0.0f;
    for (int i = lane; i < C_DIM / 4; i += 32) {
        const float4 v = l4[i];
        acc += __expf(v.x - vmax) + __expf(v.y - vmax)
             + __expf(v.z - vmax) + __expf(v.w - vmax);
    }
    #pragma unroll
    for (int off = 16; off > 0; off >>= 1) acc += __shfl_xor(acc, off, 32);

    // ---- Per-row loss term --------------------------------------------
    if (lane == 0) {
        const int   l    = label[row];
        const float sl   = lrow[l];                         // s[i, label_i]
        const float m    = xi[(size_t)l * C_DIM + imax];    // xi[label, argmax]
        const float logm = __logf(m);
        const float lse  = vmax + __logf(acc);              // log sum exp(s)
        const float denom = logm + lse;                     // log sum m*exp(s)
        vrow[wid] = logm + sl - denom;                      // log_s at label
    }
    __syncthreads();
    if (threadIdx.x == 0) {
        float t = 0.0f;
        #pragma unroll
        for (int w = 0; w < ROWS_PER_BLOCK; ++w) t += vrow[w];
        block_part[blockIdx.x] = t;
    }
}

__global__ __launch_bounds__(256) void cosen_finalize_kernel(
    const float* __restrict__ part, float* __restrict__ out, int n)
{
    __shared__ float wsum[8];
    float s = 0.0f;
    for (int i = threadIdx.x; i < n; i += 256) s += part[i];
    #pragma unroll
    for (int off = 16; off > 0; off >>= 1) s += __shfl_xor(s, off, 32);
    const int lane = threadIdx.x & 31;
    const int wid  = threadIdx.x >> 5;
    if (lane == 0) wsum[wid] = s;
    __syncthreads();
    if (threadIdx.x == 0) {
        float t = 0.0f;
        #pragma unroll
        for (int w = 0; w < 8; ++w) t += wsum[w];
        out[0] = -t / (float)B_DIM;   // NLL = -mean(log_s[i, label_i])
    }
}

extern "C" void kernel_launch(void* const* d_in, const int* in_sizes, int n_in,
                              void* d_out, int out_size, void* d_ws, size_t ws_size,
                              hipStream_t stream) {
    const float* cls   = (const float*)d_in[0];
    const int*   label = (const int*)d_in[1];   // integer input -> const int* per harness
    const float* xi    = (const float*)d_in[2];
    float* out  = (float*)d_out;
    float* part = (float*)d_ws;                 // NBLOCKS * 4 B = 16 KB scratch

    hipLaunchKernelGGL(cosen_row_kernel, dim3(NBLOCKS), dim3(THREADS), 0, stream,
                       cls, label, xi, part);
    hipLaunchKernelGGL(cosen_finalize_kernel, dim3(1), dim3(256), 0, stream,
                       part, out, NBLOCKS);
}